// HouseholderFlow_26766236188682
// MI455X (gfx1250) — compile-verified
//
#include <hip/hip_runtime.h>

typedef __attribute__((ext_vector_type(2))) float v2f;
typedef __attribute__((ext_vector_type(4))) float v4f;
typedef __attribute__((ext_vector_type(8))) float v8f;

#define DCOLS 128
#define TILE_ROWS 16

// One wave processes 16-row x 128-col tiles.
// X is loaded straight into WMMA C-layout (lane = (half<<4)+col, vgpr k = row k / k+8).
// Row dots d[m] = x_m . v  via VALU FMAs + 16-lane butterfly reduction,
// then Y = d (outer) (-2v/||v||^2) + X is done on the matrix pipe:
// 8x V_WMMA_F32_16X16X4_F32 with A[m,0]=d[m], B[0,n]=-s*v[n], C = X tile.
__global__ __launch_bounds__(256) void hh_flow_kernel(const float* __restrict__ x,
                                                      const float* __restrict__ v,
                                                      float* __restrict__ y,
                                                      int nrows) {
  const int lane = threadIdx.x & 31;
  const int low  = lane & 15;
  const int half = lane >> 4; // 0: rows 0..7 of tile, 1: rows 8..15
  const int waves_per_block = blockDim.x >> 5;
  const long wave_id = (long)blockIdx.x * waves_per_block + (threadIdx.x >> 5);
  const long nwaves  = (long)gridDim.x * waves_per_block;

  // ---- ||v||^2, computed redundantly per wave (v is 512B, L2-resident) ----
  v4f v4 = *(const v4f*)(v + lane * 4);
  float ssq = v4.x * v4.x + v4.y * v4.y + v4.z * v4.z + v4.w * v4.w;
#pragma unroll
  for (int off = 16; off >= 1; off >>= 1)
    ssq += __shfl_xor(ssq, off, 32);
  const float s = 2.0f / ssq;

  // Per-lane v values for this lane's column within each of the 8 column blocks.
  float vcol[8];
#pragma unroll
  for (int nb = 0; nb < 8; ++nb) vcol[nb] = v[nb * 16 + low];

  // B operand per column block: only row K=0 nonzero, B[0,n] = -s*v[n].
  // 4x16 B layout: K=0 row lives in lanes 0..15 of the first VGPR.
  float bbase[8];
#pragma unroll
  for (int nb = 0; nb < 8; ++nb) bbase[nb] = (lane < 16) ? (-s * vcol[nb]) : 0.0f;

  const long ntiles = (long)nrows / TILE_ROWS;
  for (long t = wave_id; t < ntiles; t += nwaves) {
    const long row0 = t * TILE_ROWS;
    const float* xt = x + row0 * DCOLS;
    float* yt = y + row0 * DCOLS;

    // Prefetch next tile for this wave (global_prefetch_b8).
    const long tn = t + nwaves;
    if (tn < ntiles) {
      const float* nx = x + tn * TILE_ROWS * DCOLS;
      __builtin_prefetch(nx + lane * 32, 0, 0);
      __builtin_prefetch(nx + 1024 + lane * 32, 0, 0);
    }

    // Load X tile in WMMA C-layout: c[nb][k] = X[8*half + k][nb*16 + low]
    float c[8][8];
#pragma unroll
    for (int k = 0; k < 8; ++k) {
      const float* rp = xt + (8 * half + k) * DCOLS + low;
#pragma unroll
      for (int nb = 0; nb < 8; ++nb)
        c[nb][k] = __builtin_nontemporal_load(rp + nb * 16);
    }

    // Per-lane dot partials for the 8 rows this half owns.
    float p[8];
#pragma unroll
    for (int k = 0; k < 8; ++k) {
      float acc = c[0][k] * vcol[0];
#pragma unroll
      for (int nb = 1; nb < 8; ++nb) acc = fmaf(c[nb][k], vcol[nb], acc);
      p[k] = acc;
    }
    // Butterfly-reduce within each 16-lane half: p[k] -> d[8*half + k] (in all lanes of the half).
#pragma unroll
    for (int k = 0; k < 8; ++k) {
#pragma unroll
      for (int off = 1; off <= 8; off <<= 1)
        p[k] += __shfl_xor(p[k], off, 32);
    }

    // Scatter d into WMMA A-layout: lane m (0..15) holds A[m,0]=d[m]; all other A slots 0.
    float a0 = 0.0f;
#pragma unroll
    for (int k = 0; k < 8; ++k) {
      const float dk  = __shfl(p[k], 0, 32);   // d[k]     (from lanes 0..15)
      const float dk8 = __shfl(p[k], 16, 32);  // d[k + 8] (from lanes 16..31)
      a0 = (lane == k)     ? dk  : a0;
      a0 = (lane == k + 8) ? dk8 : a0;
    }
    v2f A; A.x = a0; A.y = 0.0f;

    // Rank-1 update on the matrix pipe: D = A(16x4) x B(4x16) + X_tile, per column block.
#pragma unroll
    for (int nb = 0; nb < 8; ++nb) {
      v2f Bv; Bv.x = bbase[nb]; Bv.y = 0.0f;
      v8f Cv;
#pragma unroll
      for (int k = 0; k < 8; ++k) Cv[k] = c[nb][k];
      v8f Dv = __builtin_amdgcn_wmma_f32_16x16x4_f32(
          /*neg_a=*/false, A, /*neg_b=*/false, Bv,
          /*c_mod=*/(short)0, Cv, /*reuse_a=*/false, /*reuse_b=*/false);
      float* wp = yt + (8 * half) * DCOLS + nb * 16 + low;
#pragma unroll
      for (int k = 0; k < 8; ++k)
        __builtin_nontemporal_store(Dv[k], wp + k * DCOLS);
    }
  }
}

// Reference returns a tuple (y, 0): zero any trailing elements of d_out past y.
__global__ void hh_tail_zero_kernel(float* out, long begin, long end) {
  long i = begin + (long)blockIdx.x * blockDim.x + threadIdx.x;
  if (i < end) out[i] = 0.0f;
}

extern "C" void kernel_launch(void* const* d_in, const int* in_sizes, int n_in,
                              void* d_out, int out_size, void* d_ws, size_t ws_size,
                              hipStream_t stream) {
  const float* x = (const float*)d_in[0];
  const float* v = (const float*)d_in[1];
  float* y = (float*)d_out;

  const int nrows = in_sizes[0] / DCOLS;

  // 2048 blocks x 256 threads = 16384 waves; grid-stride over 131072 tiles (8 each).
  hh_flow_kernel<<<2048, 256, 0, stream>>>(x, v, y, nrows);

  const long ydone = (long)nrows * DCOLS;
  if ((long)out_size > ydone) {
    const long tail = (long)out_size - ydone;
    const int tb = 256;
    const int tg = (int)((tail + tb - 1) / tb);
    hh_tail_zero_kernel<<<tg, tb, 0, stream>>>(y, ydone, (long)out_size);
  }
}